// FNDE_44839458570639
// MI455X (gfx1250) — compile-verified
//
#include <hip/hip_runtime.h>

// ---------------------------------------------------------------------------
// FNO Neural-ODE, fully restructured into spectral space (exact algebra):
//   - one partial forward DFT of the raw input (corner modes only)
//   - 3 layers x 4 RK4 steps x 4 f-evals, each f-eval = per-mode CxC complex
//     matmul via V_WMMA_F32_16X16X4_F32 + bias at mode(0,0) + Pi-symmetrization
//     of the ky=0 column (models rfft2(irfft2(.)) roundtrip exactly)
//   - projection folded to a channel contraction in spectral space
//   - one partial inverse DFT producing the final [16,1,128,128] output
// Working set ~70MB in d_ws, L2-resident on MI455X.
// ---------------------------------------------------------------------------

typedef float v2f __attribute__((ext_vector_type(2)));
typedef float v8f __attribute__((ext_vector_type(8)));

#define Bb    16
#define Cc    64
#define Dd    128
#define Mm    16
#define Ll    3
#define NMODE 512                      // 2 corners * 16 kx * 16 ky
#define NSPEC (NMODE*Cc*Bb)            // 524288 complex-component elems per plane
#define WLAYER (2*NMODE*Cc*Cc)         // floats per packed layer (re+im planes)

// ---------------- twiddle tables ----------------
__global__ void fnde_build_tables(float* cs, float* sn) {
    int t = threadIdx.x;               // 128 threads
    float ang = (float)(6.283185307179586 / 128.0) * (float)t;
    cs[t] = cosf(ang);
    sn[t] = sinf(ang);
}

// ---------------- pack spectral weights: W[l][mode][o][i] (re plane, im plane)
// einsum "bixy,ioxy->boxy"  =>  W[o][i] = w{1,2}[i][o][mx][ky]
__global__ void fnde_pack_weights(const float* __restrict__ w1,
                                  const float* __restrict__ w2,
                                  float* __restrict__ Wp) {
    int idx = blockIdx.x * 256 + threadIdx.x;   // L*NMODE*Cc*Cc = 6291456
    int l   = idx / (NMODE * Cc * Cc);
    int rem = idx % (NMODE * Cc * Cc);
    int m   = rem >> 12;                        // mode
    int oi  = rem & 4095;
    int o   = oi >> 6;
    int i   = oi & 63;
    int corner = m >> 8;
    int mx = (m >> 4) & 15;
    int ky = m & 15;
    const float* src = corner ? w2 : w1;
    int sidx = ((((l * Cc + i) * Cc + o) * Mm + mx) * Mm + ky) * 2;
    float* dst = Wp + (size_t)l * WLAYER + (size_t)m * 4096 + oi;
    dst[0]              = src[sidx];            // re
    dst[NMODE * Cc * Cc] = src[sidx + 1];       // im plane offset
}

// ---------------- forward partial DFT of raw input z[b,0,x,y] ----------------
// stage 1: over y -> Ay[b][x][ky] complex
__global__ void fnde_fwd_dft1(const float* __restrict__ z,
                              const float* __restrict__ cs,
                              const float* __restrict__ sn,
                              float* __restrict__ Ay) {
    int t  = blockIdx.x * 256 + threadIdx.x;    // Bb*Dd*Mm = 32768
    int b  = t >> 11;
    int x  = (t >> 4) & 127;
    int ky = t & 15;
    const float* row = z + (b * 128 + x) * 128;
    float re = 0.f, im = 0.f;
    for (int y = 0; y < 128; ++y) {
        float v = row[y];
        int a = (ky * y) & 127;
        re += v * cs[a];
        im -= v * sn[a];
    }
    Ay[t * 2]     = re;
    Ay[t * 2 + 1] = im;
}

// stage 2: over x -> Zhat[b][mode] complex (interleaved)
__global__ void fnde_fwd_dft2(const float* __restrict__ Ay,
                              const float* __restrict__ cs,
                              const float* __restrict__ sn,
                              float* __restrict__ Zhat) {
    int t = blockIdx.x * 256 + threadIdx.x;     // Bb*NMODE = 8192
    int b = t >> 9;
    int m = t & 511;
    int corner = m >> 8, mx = (m >> 4) & 15, ky = m & 15;
    int kx = corner ? (112 + mx) : mx;
    float re = 0.f, im = 0.f;
    for (int x = 0; x < 128; ++x) {
        int base = ((b * 128 + x) * 16 + ky) * 2;
        float ar = Ay[base], ai = Ay[base + 1];
        int a = (kx * x) & 127;
        float c = cs[a], s = sn[a];
        re += ar * c + ai * s;                  // * e^{-i theta}
        im += ai * c - ar * s;
    }
    Zhat[t * 2]     = re;
    Zhat[t * 2 + 1] = im;
}

// ---------------- lift into spectral state Z[mode][c][b]; zero S_total -------
__global__ void fnde_init_spec(const float* __restrict__ Zhat,
                               const float* __restrict__ lw,
                               const float* __restrict__ lb,
                               float* __restrict__ Zre, float* __restrict__ Zim,
                               float* __restrict__ STre, float* __restrict__ STim) {
    int e = blockIdx.x * 256 + threadIdx.x;     // NSPEC
    int m = e >> 10;
    int c = (e >> 4) & 63;
    int b = e & 15;
    float zr = Zhat[(b * 512 + m) * 2];
    float zi = Zhat[(b * 512 + m) * 2 + 1];
    float w = lw[c];
    Zre[e] = w * zr + ((m == 0) ? lb[c] * 16384.0f : 0.0f);  // bias -> mode(0,0)*D^2
    Zim[e] = w * zi;
    STre[e] = 0.0f;
    STim[e] = 0.0f;
}

// ---------------- G-apply: K[mode] = W[mode] * V[mode] (+ bias*D^2 at mode 0)
// One wave per mode. Complex 64x64 @ 64x16 via V_WMMA_F32_16X16X4_F32.
__global__ __launch_bounds__(128)
void fnde_gapply(const float* __restrict__ Vre, const float* __restrict__ Vim,
                 const float* __restrict__ Wp,   // layer base (re plane; im at +NMODE*Cc*Cc)
                 float* __restrict__ Kre, float* __restrict__ Kim,
                 const float* __restrict__ bias) {
    int tid  = blockIdx.x * 128 + threadIdx.x;
    int mode = tid >> 5;                        // 512 waves total
    int lane = threadIdx.x & 31;
    int laneHi = lane >> 4;                     // 0/1
    int laneLo = lane & 15;

    const float* Wr = Wp + (size_t)mode * 4096;
    const float* Wi = Wr + (size_t)NMODE * Cc * Cc;
    const float* Vr = Vre + mode * 1024;
    const float* Vi = Vim + mode * 1024;

    v8f cr0 = {}, cr1 = {}, cr2 = {}, cr3 = {};
    v8f ci0 = {}, ci1 = {}, ci2 = {}, ci3 = {};

    const int bOff = (laneHi << 5) + laneLo;    // B operand address in V plane
    const int aOff = laneLo * 64 + laneHi * 2;  // A operand address in W plane

#pragma unroll 4
    for (int kk = 0; kk < 16; ++kk) {
        int vb = kk * 64 + bOff;
        v2f br, bi, bin;
        br.x = Vr[vb];      br.y = Vr[vb + 16];
        bi.x = Vi[vb];      bi.y = Vi[vb + 16];
        bin.x = -bi.x;      bin.y = -bi.y;      // for Cr = Wr*Xr - Wi*Xi
        int ab = aOff + kk * 4;

#define FNDE_MT(MT, CR, CI)                                                          \
        {                                                                            \
            v2f ar = *(const v2f*)(Wr + ab + MT * 1024);                             \
            v2f ai = *(const v2f*)(Wi + ab + MT * 1024);                             \
            CR = __builtin_amdgcn_wmma_f32_16x16x4_f32(false, ar, false, br,         \
                                                       (short)0, CR, false, false);  \
            CR = __builtin_amdgcn_wmma_f32_16x16x4_f32(false, ai, false, bin,        \
                                                       (short)0, CR, false, false);  \
            CI = __builtin_amdgcn_wmma_f32_16x16x4_f32(false, ar, false, bi,         \
                                                       (short)0, CI, false, false);  \
            CI = __builtin_amdgcn_wmma_f32_16x16x4_f32(false, ai, false, br,         \
                                                       (short)0, CI, false, false);  \
        }
        FNDE_MT(0, cr0, ci0)
        FNDE_MT(1, cr1, ci1)
        FNDE_MT(2, cr2, ci2)
        FNDE_MT(3, cr3, ci3)
#undef FNDE_MT
    }

    float* Kor = Kre + mode * 1024;
    float* Koi = Kim + mode * 1024;
    bool addBias = (mode == 0);

#define FNDE_ST(MT, CR, CI)                                                      \
    {                                                                            \
        int obase = MT * 16 + (laneHi << 3);                                     \
        _Pragma("unroll")                                                        \
        for (int j = 0; j < 8; ++j) {                                            \
            float vr = CR[j];                                                    \
            if (addBias) vr += bias[obase + j] * 16384.0f;                       \
            Kor[(obase + j) * 16 + laneLo] = vr;                                 \
            Koi[(obase + j) * 16 + laneLo] = CI[j];                              \
        }                                                                        \
    }
    FNDE_ST(0, cr0, ci0)
    FNDE_ST(1, cr1, ci1)
    FNDE_ST(2, cr2, ci2)
    FNDE_ST(3, cr3, ci3)
#undef FNDE_ST
}

// ---------------- Pi projection (rfft2 o irfft2 roundtrip at ky==0) ----------
__device__ inline void fnde_pi(const float* __restrict__ Re, const float* __restrict__ Im,
                               int e, float kr, float ki, float& pr, float& pim) {
    int m = e >> 10;
    int ky = m & 15;
    if (ky != 0) { pr = kr; pim = ki; return; }
    int corner = m >> 8;
    int mx = (m >> 4) & 15;
    if (corner == 1 && mx == 0) { pr = 0.5f * kr; pim = 0.5f * ki; return; }  // kx=112, partner kx=16 truncated
    int pm = (mx == 0) ? m : ((corner == 0) ? (256 + (16 - mx) * 16) : ((16 - mx) * 16));
    int pe = (pm << 10) + (e & 1023);
    pr  = 0.5f * (kr + Re[pe]);
    pim = 0.5f * (ki - Im[pe]);
}

// ---------------- RK4 sub-step combine (phases 1..3) -------------------------
__global__ void fnde_spec_mid(const float* __restrict__ Zre, const float* __restrict__ Zim,
                              const float* __restrict__ Kre, const float* __restrict__ Kim,
                              float* __restrict__ Ure, float* __restrict__ Uim,
                              float* __restrict__ SSre, float* __restrict__ SSim,
                              const float* __restrict__ ts, int s, int phase) {
    int e = blockIdx.x * 256 + threadIdx.x;
    float h  = ts[s + 1] - ts[s];
    float uc = (phase == 3) ? h : 0.5f * h;
    float kr = Kre[e], ki = Kim[e];
    if (phase == 1) { SSre[e] = kr;          SSim[e] = ki; }
    else            { SSre[e] += 2.0f * kr;  SSim[e] += 2.0f * ki; }
    float pr, pim;
    fnde_pi(Kre, Kim, e, kr, ki, pr, pim);
    Ure[e] = Zre[e] + uc * pr;
    Uim[e] = Zim[e] + uc * pim;
}

// ---------------- RK4 final combine: Z += h/6*Pi(S4); S_total += h/6*S4 ------
__global__ void fnde_spec_fin(float* __restrict__ Zre, float* __restrict__ Zim,
                              const float* __restrict__ Kre, const float* __restrict__ Kim,
                              const float* __restrict__ SSre, const float* __restrict__ SSim,
                              float* __restrict__ STre, float* __restrict__ STim,
                              const float* __restrict__ ts, int s) {
    int e = blockIdx.x * 256 + threadIdx.x;
    float h6 = (ts[s + 1] - ts[s]) * (1.0f / 6.0f);
    float s4r = SSre[e] + Kre[e];
    float s4i = SSim[e] + Kim[e];
    int m = e >> 10;
    int ky = m & 15;
    float pr, pim;
    if (ky != 0) { pr = s4r; pim = s4i; }
    else {
        int corner = m >> 8, mx = (m >> 4) & 15;
        if (corner == 1 && mx == 0) { pr = 0.5f * s4r; pim = 0.5f * s4i; }
        else {
            int pm = (mx == 0) ? m : ((corner == 0) ? (256 + (16 - mx) * 16) : ((16 - mx) * 16));
            int pe = (pm << 10) + (e & 1023);
            float qr = SSre[pe] + Kre[pe];
            float qi = SSim[pe] + Kim[pe];
            pr  = 0.5f * (s4r + qr);
            pim = 0.5f * (s4i - qi);
        }
    }
    Zre[e] += h6 * pr;
    Zim[e] += h6 * pim;
    STre[e] += h6 * s4r;
    STim[e] += h6 * s4i;
}

// ---------------- projection prep: q[c], alpha, beta -------------------------
__global__ void fnde_proj_prep(const float* __restrict__ p1w, const float* __restrict__ p1b,
                               const float* __restrict__ p2w, const float* __restrict__ p2b,
                               const float* __restrict__ lw,  const float* __restrict__ lb,
                               float* __restrict__ q, float* __restrict__ scal) {
    __shared__ float qs[64];
    int c = threadIdx.x;                        // 64 threads
    float acc = 0.f;
    for (int p = 0; p < 64; ++p) acc += p2w[p] * p1w[p * 64 + c];
    qs[c] = acc;
    q[c] = acc;
    __syncthreads();
    if (c == 0) {
        float r = p2b[0];
        for (int p = 0; p < 64; ++p) r += p2w[p] * p1b[p];
        float al = 0.f, be = r;
        for (int i = 0; i < 64; ++i) { al += qs[i] * lw[i]; be += qs[i] * lb[i]; }
        scal[0] = al;
        scal[1] = be;
    }
}

// ---------------- contract channels in spectral space: T[b][mode] -----------
__global__ void fnde_proj_modes(const float* __restrict__ STre, const float* __restrict__ STim,
                                const float* __restrict__ q, float* __restrict__ T) {
    int t = blockIdx.x * 256 + threadIdx.x;     // Bb*NMODE = 8192
    int b = t >> 9;
    int m = t & 511;
    const float* pr = STre + m * 1024 + b;
    const float* pi = STim + m * 1024 + b;
    float re = 0.f, im = 0.f;
    for (int c = 0; c < 64; ++c) {
        float w = q[c];
        re += w * pr[c * 16];
        im += w * pi[c * 16];
    }
    T[t * 2]     = re;
    T[t * 2 + 1] = im;
}

// ---------------- inverse partial DFT stage 1 (over kx) ----------------------
__global__ void fnde_inv_dft1(const float* __restrict__ T,
                              const float* __restrict__ cs, const float* __restrict__ sn,
                              float* __restrict__ G) {
    int t  = blockIdx.x * 256 + threadIdx.x;    // Bb*Dd*Mm = 32768
    int b  = t >> 11;
    int x  = (t >> 4) & 127;
    int ky = t & 15;
    float re = 0.f, im = 0.f;
    for (int corner = 0; corner < 2; ++corner) {
        for (int mx = 0; mx < 16; ++mx) {
            int m  = corner * 256 + mx * 16 + ky;
            int kx = corner ? (112 + mx) : mx;
            float tr = T[(b * 512 + m) * 2];
            float ti = T[(b * 512 + m) * 2 + 1];
            int a = (kx * x) & 127;
            float c = cs[a], s = sn[a];
            re += tr * c - ti * s;              // * e^{+i theta}
            im += tr * s + ti * c;
        }
    }
    G[t * 2]     = re;
    G[t * 2 + 1] = im;
}

// ---------------- inverse stage 2 (over ky, irfft semantics) + output --------
__global__ void fnde_inv_dft2_out(const float* __restrict__ G, const float* __restrict__ z,
                                  const float* __restrict__ cs, const float* __restrict__ sn,
                                  const float* __restrict__ scal, float* __restrict__ out) {
    int t = blockIdx.x * 256 + threadIdx.x;     // Bb*Dd*Dd = 262144
    int b = t >> 14;
    int x = (t >> 7) & 127;
    int y = t & 127;
    const float* g = G + ((b * 128 + x) * 16) * 2;
    float acc = g[0];                           // ky=0: real part only (irfft drops Im)
    for (int ky = 1; ky < 16; ++ky) {
        int a = (ky * y) & 127;
        acc += 2.0f * (g[ky * 2] * cs[a] - g[ky * 2 + 1] * sn[a]);
    }
    out[t] = scal[0] * z[t] + scal[1] + acc * (1.0f / 16384.0f);
}

// ---------------------------------------------------------------------------
extern "C" void kernel_launch(void* const* d_in, const int* in_sizes, int n_in,
                              void* d_out, int out_size, void* d_ws, size_t ws_size,
                              hipStream_t stream) {
    (void)in_sizes; (void)n_in; (void)out_size; (void)ws_size;

    const float* zin    = (const float*)d_in[0];
    const float* lift_w = (const float*)d_in[1];
    const float* lift_b = (const float*)d_in[2];
    const float* spec_w1 = (const float*)d_in[3];
    const float* spec_w2 = (const float*)d_in[4];
    const float* fl_bias = (const float*)d_in[5];
    const float* p1_w = (const float*)d_in[6];
    const float* p1_b = (const float*)d_in[7];
    const float* p2_w = (const float*)d_in[8];
    const float* p2_b = (const float*)d_in[9];
    const float* ts   = (const float*)d_in[10];
    float* outp = (float*)d_out;

    float* w = (float*)d_ws;
    float* Wp   = w;  w += (size_t)Ll * WLAYER;     // 12.58M floats
    float* Zre  = w;  w += NSPEC;  float* Zim  = w;  w += NSPEC;
    float* Ure  = w;  w += NSPEC;  float* Uim  = w;  w += NSPEC;
    float* Kre  = w;  w += NSPEC;  float* Kim  = w;  w += NSPEC;
    float* SSre = w;  w += NSPEC;  float* SSim = w;  w += NSPEC;
    float* STre = w;  w += NSPEC;  float* STim = w;  w += NSPEC;
    float* Zhat = w;  w += Bb * NMODE * 2;
    float* Ay   = w;  w += Bb * Dd * Mm * 2;
    float* Tm   = w;  w += Bb * NMODE * 2;
    float* Gi   = w;  w += Bb * Dd * Mm * 2;
    float* qv   = w;  w += 64;
    float* cs   = w;  w += 128;
    float* sn   = w;  w += 128;
    float* scal = w;  w += 8;

    fnde_build_tables<<<1, 128, 0, stream>>>(cs, sn);
    fnde_pack_weights<<<(Ll * NMODE * Cc * Cc) / 256, 256, 0, stream>>>(spec_w1, spec_w2, Wp);
    fnde_fwd_dft1<<<(Bb * Dd * Mm) / 256, 256, 0, stream>>>(zin, cs, sn, Ay);
    fnde_fwd_dft2<<<(Bb * NMODE) / 256, 256, 0, stream>>>(Ay, cs, sn, Zhat);
    fnde_init_spec<<<NSPEC / 256, 256, 0, stream>>>(Zhat, lift_w, lift_b, Zre, Zim, STre, STim);

    for (int l = 0; l < Ll; ++l) {
        const float* Wl = Wp + (size_t)l * WLAYER;
        const float* bl = fl_bias + l * Cc;
        for (int s = 0; s < 4; ++s) {
            fnde_gapply<<<128, 128, 0, stream>>>(Zre, Zim, Wl, Kre, Kim, bl);
            fnde_spec_mid<<<NSPEC / 256, 256, 0, stream>>>(Zre, Zim, Kre, Kim, Ure, Uim,
                                                           SSre, SSim, ts, s, 1);
            fnde_gapply<<<128, 128, 0, stream>>>(Ure, Uim, Wl, Kre, Kim, bl);
            fnde_spec_mid<<<NSPEC / 256, 256, 0, stream>>>(Zre, Zim, Kre, Kim, Ure, Uim,
                                                           SSre, SSim, ts, s, 2);
            fnde_gapply<<<128, 128, 0, stream>>>(Ure, Uim, Wl, Kre, Kim, bl);
            fnde_spec_mid<<<NSPEC / 256, 256, 0, stream>>>(Zre, Zim, Kre, Kim, Ure, Uim,
                                                           SSre, SSim, ts, s, 3);
            fnde_gapply<<<128, 128, 0, stream>>>(Ure, Uim, Wl, Kre, Kim, bl);
            fnde_spec_fin<<<NSPEC / 256, 256, 0, stream>>>(Zre, Zim, Kre, Kim, SSre, SSim,
                                                           STre, STim, ts, s);
        }
    }

    fnde_proj_prep<<<1, 64, 0, stream>>>(p1_w, p1_b, p2_w, p2_b, lift_w, lift_b, qv, scal);
    fnde_proj_modes<<<(Bb * NMODE) / 256, 256, 0, stream>>>(STre, STim, qv, Tm);
    fnde_inv_dft1<<<(Bb * Dd * Mm) / 256, 256, 0, stream>>>(Tm, cs, sn, Gi);
    fnde_inv_dft2_out<<<(Bb * Dd * Dd) / 256, 256, 0, stream>>>(Gi, zin, cs, sn, scal, outp);
}